// Upsample_62543313764584
// MI455X (gfx1250) — compile-verified
//
#include <hip/hip_runtime.h>

// Problem constants from the reference (out_h/out_w inputs are fixed 512).
#define B_SZ   16
#define CH     8
#define H_IN   64
#define W_IN   64
#define H_OUT  512
#define W_OUT  512
#define ROWS_PER_STRIPE 64
#define STRIPES (H_OUT / ROWS_PER_STRIPE)

typedef unsigned int v4u __attribute__((ext_vector_type(4)));
typedef int          v8i __attribute__((ext_vector_type(8)));
typedef int          v4i __attribute__((ext_vector_type(4)));

// Replicates reference _axis_index_weights for one coordinate.
// t = linspace(0,1,n_out)[o]; coord = t*d; idx = clip(coord/d*(n_in-1), 0, n_in-1)
__device__ __forceinline__ void axis_iw(float t, float d, int nin,
                                        int& i0, float& w) {
    float coord = t * d;
    float idx   = (coord / d) * (float)(nin - 1);
    idx = fminf(fmaxf(idx, 0.0f), (float)(nin - 1));
    int i = (int)floorf(idx);
    i = min(i, nin - 2);
    i0 = i;
    w  = idx - (float)i;
}

__global__ __launch_bounds__(256) void Upsample_62543313764584_kernel(
    const float* __restrict__ inp,          // (B, C, 64, 64)
    const float* __restrict__ domain_sizes, // (B, 2)
    float* __restrict__ out)                // (B, C, 512, 512)
{
    __shared__ float tile[H_IN * W_IN];     // 16 KB staged input plane
    __shared__ int2  vtab[ROWS_PER_STRIPE]; // per-output-row {i0h, bits(wh)}

    const int plane  = blockIdx.y;          // b*C + c  in [0,128)
    const int stripe = blockIdx.x;          // [0,8)
    const int b      = plane / CH;
    const float d0   = domain_sizes[2 * b + 0];
    const float d1   = domain_sizes[2 * b + 1];

    const float* src = inp + (size_t)plane * (H_IN * W_IN);
    const int oy_base = stripe * ROWS_PER_STRIPE;

    // ---- Stage the 16KB input plane into LDS via the Tensor Data Mover ----
#if __has_builtin(__builtin_amdgcn_tensor_load_to_lds) && \
    __has_builtin(__builtin_amdgcn_s_wait_tensorcnt)
    if (threadIdx.x < 32) {                 // wave 0 only issues the DMA
        unsigned long long ga = (unsigned long long)(uintptr_t)src;
        unsigned int lds_lo   = (unsigned int)(uintptr_t)(&tile[0]);

        // D# group 0 (128b): count=1 | lds_addr | global_addr[56:0] | type=2
        v4u g0;
        g0.x = 1u;                                        // count=1, user mode
        g0.y = lds_lo;                                    // lds_addr (bytes)
        g0.z = (unsigned int)(ga & 0xFFFFFFFFu);          // global_addr[31:0]
        g0.w = (unsigned int)((ga >> 32) & 0x01FFFFFFu)   // global_addr[56:32]
             | (2u << 30);                                // type = 2 ("image")

        // D# group 1 (256b): 2D tensor, data_size=4B, 64x64 tile == tensor
        v8i g1;
        g1[0] = (int)(2u << 16);        // workgroup_mask=0, data_size=2 (4B)
        g1[1] = (int)(64u << 16);       // tensor_dim0 = 64      (bits 79:48 lo)
        g1[2] = (int)(64u << 16);       // tensor_dim1 = 64      (bits 111:80 lo)
        g1[3] = (int)(64u << 16);       // tile_dim0   = 64      (bits 127:112)
        g1[4] = (int)(64u);             // tile_dim1=64, tile_dim2=0
        g1[5] = (int)(64u);             // tensor_dim0_stride = 64
        g1[6] = (int)(4096u << 16);     // tensor_dim1_stride lo16 (unused for 2D)
        g1[7] = 0;                      // tensor_dim1_stride hi
        v4i z4 = {0, 0, 0, 0};          // groups 2/3 unused for 2D tile
        v8i z8 = {0, 0, 0, 0, 0, 0, 0, 0};

        __builtin_amdgcn_tensor_load_to_lds(g0, g1, z4, z4, z8, 0);
        // NOTE: no wait yet — overlap DMA with weight-table precompute below.
    }
#else
    // Fallback: cooperative b128 loads into LDS
    for (int i = threadIdx.x; i < (H_IN * W_IN) / 4; i += 256) {
        ((float4*)tile)[i] = ((const float4*)src)[i];
    }
#endif

    // ---- Precompute vertical (i0h, wh) for this stripe's 64 rows (overlaps DMA) ----
    if (threadIdx.x < ROWS_PER_STRIPE) {
        const int oy = oy_base + (int)threadIdx.x;
        int   ih;
        float wh;
        float t = (float)oy * (1.0f / (float)(H_OUT - 1));
        axis_iw(t, d0, H_IN, ih, wh);
        vtab[threadIdx.x] = make_int2(ih, __float_as_int(wh));
    }

#if __has_builtin(__builtin_amdgcn_tensor_load_to_lds) && \
    __has_builtin(__builtin_amdgcn_s_wait_tensorcnt)
    if (threadIdx.x < 32) {
        __builtin_amdgcn_s_wait_tensorcnt(0);   // wave 0 drains the TDM op
    }
#endif
    __syncthreads();

    // ---- Interpolate: each thread owns 4 consecutive output columns ----
    const int qx   = threadIdx.x & 127;     // float4 column index [0,128)
    const int half = threadIdx.x >> 7;      // row phase 0/1
    const int ox0  = qx * 4;

    int   iw[4];
    float ww[4];
#pragma unroll
    for (int j = 0; j < 4; ++j) {
        float t = (float)(ox0 + j) * (1.0f / (float)(W_OUT - 1));
        axis_iw(t, d1, W_IN, iw[j], ww[j]);
    }

    float* oplane = out + (size_t)plane * (H_OUT * W_OUT);

#pragma unroll 2
    for (int r = 0; r < ROWS_PER_STRIPE / 2; ++r) {
        const int r2 = half + r * 2;        // local row [0,64)
        const int oy = oy_base + r2;

        const int2 v  = vtab[r2];           // single ds_load_b64
        const int  ih = v.x;
        const float wh = __int_as_float(v.y);

        const float* r0p = &tile[ih * W_IN];
        const float* r1p = r0p + W_IN;

        float4 res;
        float* rp = (float*)&res;
#pragma unroll
        for (int j = 0; j < 4; ++j) {
            float x00 = r0p[iw[j]];
            float x01 = r0p[iw[j] + 1];
            float x10 = r1p[iw[j]];
            float x11 = r1p[iw[j] + 1];
            float top = fmaf(ww[j], x01 - x00, x00);
            float bot = fmaf(ww[j], x11 - x10, x10);
            rp[j]     = fmaf(wh,    bot - top, top);
        }
        *(float4*)(&oplane[(size_t)oy * W_OUT + ox0]) = res;  // coalesced b128
    }
}

extern "C" void kernel_launch(void* const* d_in, const int* in_sizes, int n_in,
                              void* d_out, int out_size, void* d_ws, size_t ws_size,
                              hipStream_t stream) {
    const float* inp = (const float*)d_in[0];   // (16,8,64,64) f32
    const float* ds  = (const float*)d_in[1];   // (16,2) f32
    // d_in[2]=out_h(512), d_in[3]=out_w(512): compile-time constants here.
    float* out = (float*)d_out;

    dim3 grid(STRIPES, B_SZ * CH);              // 8 x 128 blocks
    Upsample_62543313764584_kernel<<<grid, 256, 0, stream>>>(inp, ds, out);
}